// Q2VRankerStage2_45079976739056
// MI455X (gfx1250) — compile-verified
//
#include <hip/hip_runtime.h>
#include <hip/hip_bf16.h>

typedef __attribute__((ext_vector_type(2))) float v2f;
typedef __attribute__((ext_vector_type(8))) float v8f;

#define WMMA_F32(a, b, c) \
    __builtin_amdgcn_wmma_f32_16x16x4_f32(false, (a), false, (b), (short)0, (c), false, false)

// ---------------------------------------------------------------------------
// Kernel 1: qn = l2norm(qfeat @ qfc_w.T + qfc_b)   (B=32 blocks, 256 threads)
// ---------------------------------------------------------------------------
__global__ void qn_kernel(const float* __restrict__ qfeat,
                          const float* __restrict__ w,
                          const float* __restrict__ bias,
                          float* __restrict__ qn) {
    __shared__ float sx[512];
    __shared__ float sq[512];
    __shared__ float sred[256];
    const int b = blockIdx.x;
    const int t = threadIdx.x;

    sx[t]       = qfeat[b * 512 + t];
    sx[t + 256] = qfeat[b * 512 + t + 256];
    __syncthreads();

    for (int rep = 0; rep < 2; ++rep) {
        const int j = t + rep * 256;
        float acc = bias[j];
        const float* wr = w + (size_t)j * 512;
        for (int d = 0; d < 512; ++d) acc = fmaf(sx[d], wr[d], acc);
        sq[j] = acc;
    }
    __syncthreads();

    sred[t] = sq[t] * sq[t] + sq[t + 256] * sq[t + 256];
    __syncthreads();
    for (int off = 128; off > 0; off >>= 1) {
        if (t < off) sred[t] += sred[t + off];
        __syncthreads();
    }
    const float inv = 1.0f / fmaxf(sqrtf(sred[0]), 1e-12f);
    qn[b * 512 + t]       = sq[t] * inv;
    qn[b * 512 + t + 256] = sq[t + 256] * inv;
}

// ---------------------------------------------------------------------------
// Kernel 2: per-(b,k) window: gather + posemb, 8-head attention (f32 WMMA),
// residual, feats_out store, and score vs qn. One block per window.
// ---------------------------------------------------------------------------
__global__ void encode_kernel(const float* __restrict__ vfeats,   // (1600,512)
                              const float* __restrict__ pos_emb,  // (400,512)
                              const float* __restrict__ wq, const float* __restrict__ biasq,
                              const float* __restrict__ wk, const float* __restrict__ biask,
                              const float* __restrict__ wv, const float* __restrict__ biasv,
                              const int*   __restrict__ hits,       // (32,16)
                              const float* __restrict__ ctx_scores, // (32,A)
                              const float* __restrict__ qn,         // (32,512)
                              float* __restrict__ out_feat,         // (512,snip,512)
                              float* __restrict__ out_merge,        // (512)
                              float* __restrict__ out_sc,           // (512)
                              int snip, int A) {
    extern __shared__ float smem[];
    const int ST16 = ((snip + 15) >> 4) << 4;   // rows padded to multiple of 16
    const int SP   = ST16 + 4;                  // score-row stride (even)
    float* X   = smem;                          // ST16 x 516
    float* QH  = X   + ST16 * 516;              // ST16 x 68
    float* KH  = QH  + ST16 * 68;
    float* VH  = KH  + ST16 * 68;
    float* P   = VH  + ST16 * 68;               // ST16 x SP
    float* SQN = P   + ST16 * SP;               // 512
    float* SSS = SQN + 512;                     // ST16
    float* SDQ = SSS + ST16;                    // ST16
    float* SRED = SDQ + ST16;                   // 256

    const int tid  = threadIdx.x;
    const int lane = tid & 31;
    const int wave = tid >> 5;
    const int NW   = 8;
    const int m_l  = lane & 15;
    const int l16  = lane >> 4;

    const int bkid = blockIdx.x;
    const int b    = bkid >> 4;
    const int anchor = hits[bkid];
    const float* src = vfeats + (size_t)anchor * snip * 512;

    // ---- stage x = feats + relu(pos_emb); zero pad rows; load qn row ----
    for (int i = tid; i < ST16 * 512; i += 256) {
        const int s = i >> 9, d = i & 511;
        float v = 0.0f;
        if (s < snip) v = src[s * 512 + d] + fmaxf(pos_emb[s * 512 + d], 0.0f);
        X[s * 516 + d] = v;
    }
    for (int d = tid; d < 512; d += 256) SQN[d] = qn[b * 512 + d];
    for (int s = tid; s < ST16; s += 256) { SSS[s] = 0.0f; SDQ[s] = 0.0f; }
    __syncthreads();

    const int MT = ST16 >> 4;

    for (int h = 0; h < 8; ++h) {
        const int hbase = h * 64;

        // ---- Q/K/V projections: C(ST16 x 64) = X(ST16x512) * W_h^T + bias ----
        const int qkv_tiles = 3 * MT * 4;
        for (int t = wave; t < qkv_tiles; t += NW) {
            const int mat = t % 3;
            const int rem = t / 3;
            const int mt = rem >> 2, nt = rem & 3;
            const float* W; const float* Bv; float* Dst;
            if (mat == 0)      { W = wq; Bv = biasq; Dst = QH; }
            else if (mat == 1) { W = wk; Bv = biask; Dst = KH; }
            else               { W = wv; Bv = biasv; Dst = VH; }
            const int wrow_idx = hbase + nt * 16 + m_l;     // output feature (B col)
            const float* wrow = W + (size_t)wrow_idx * 512;
            const float* xrow = X + (mt * 16 + m_l) * 516;
            v8f c = {};
            for (int k0 = 0; k0 < 512; k0 += 4) {
                const int kk = k0 + 2 * l16;
                const v2f a  = *(const v2f*)(xrow + kk);
                const v2f bb = *(const v2f*)(wrow + kk);
                c = WMMA_F32(a, bb, c);
            }
            const float bias = Bv[wrow_idx];
            const int col = nt * 16 + m_l;
            for (int v = 0; v < 8; ++v) {
                const int row = mt * 16 + v + 8 * l16;
                Dst[row * 68 + col] = c[v] + bias;
            }
        }
        __syncthreads();

        // ---- scores P = (Q K^T) / 8 ----
        const int s_tiles = MT * MT;
        for (int t = wave; t < s_tiles; t += NW) {
            const int mt = t / MT, nt = t % MT;
            const float* qrow = QH + (mt * 16 + m_l) * 68;
            const float* krow = KH + (nt * 16 + m_l) * 68;   // B[k][n] = KH[n][k]
            v8f c = {};
            for (int k0 = 0; k0 < 64; k0 += 4) {
                const int kk = k0 + 2 * l16;
                const v2f a  = *(const v2f*)(qrow + kk);
                const v2f bb = *(const v2f*)(krow + kk);
                c = WMMA_F32(a, bb, c);
            }
            for (int v = 0; v < 8; ++v) {
                const int row = mt * 16 + v + 8 * l16;
                P[row * SP + nt * 16 + m_l] = c[v] * 0.125f;
            }
        }
        __syncthreads();

        // ---- softmax rows (one thread per row), zero padded columns ----
        if (tid < snip) {
            float* prow = P + tid * SP;
            float mx = -INFINITY;
            for (int j = 0; j < snip; ++j) mx = fmaxf(mx, prow[j]);
            float sum = 0.0f;
            for (int j = 0; j < snip; ++j) { const float e = __expf(prow[j] - mx); prow[j] = e; sum += e; }
            const float inv = 1.0f / sum;
            for (int j = 0; j < snip; ++j) prow[j] *= inv;
            for (int j = snip; j < ST16; ++j) prow[j] = 0.0f;
        }
        __syncthreads();

        // ---- ctx = P @ V ; residual; store feats_out; accumulate scoring ----
        const int c_tiles = MT * 4;
        for (int t = wave; t < c_tiles; t += NW) {
            const int mt = t >> 2, nt = t & 3;
            const float* prow = P + (mt * 16 + m_l) * SP;
            v8f c = {};
            for (int k0 = 0; k0 < ST16; k0 += 4) {
                const int kk = k0 + 2 * l16;
                const v2f a = *(const v2f*)(prow + kk);
                v2f bb = { VH[kk * 68 + nt * 16 + m_l], VH[(kk + 1) * 68 + nt * 16 + m_l] };
                c = WMMA_F32(a, bb, c);
            }
            const int col = hbase + nt * 16 + m_l;
            const float qv = SQN[col];
            for (int v = 0; v < 8; ++v) {
                const int row = mt * 16 + v + 8 * l16;
                if (row < snip) {
                    const float o = src[row * 512 + col] + c[v];
                    out_feat[((size_t)bkid * snip + row) * 512 + col] = o;
                    atomicAdd(&SSS[row], o * o);
                    atomicAdd(&SDQ[row], o * qv);
                }
            }
        }
        __syncthreads();
    }

    // ---- scores: sc = mean_s( dot(row, qn) / max(||row||, eps) ) ----
    float own = 0.0f;
    if (tid < snip) own = SDQ[tid] / fmaxf(sqrtf(SSS[tid]), 1e-12f);
    SRED[tid] = own;
    __syncthreads();
    for (int off = 128; off > 0; off >>= 1) {
        if (tid < off) SRED[tid] += SRED[tid + off];
        __syncthreads();
    }
    if (tid == 0) {
        const float sc = SRED[0] / (float)snip;
        out_sc[bkid]    = sc;
        out_merge[bkid] = ctx_scores[b * A + anchor] + sc;
    }
}

// ---------------------------------------------------------------------------
extern "C" void kernel_launch(void* const* d_in, const int* in_sizes, int n_in,
                              void* d_out, int out_size, void* d_ws, size_t ws_size,
                              hipStream_t stream) {
    (void)in_sizes; (void)n_in; (void)out_size; (void)ws_size;
    // setup_inputs() order: vfeats, qfeat, (qv_ctx_scores_i, hit_indices_i) x4,
    //                       qfc_w, qfc_b, pos_emb, wq, bq, wk, bk, wv, bv
    const float* vfeats = (const float*)d_in[0];
    const float* qfeat  = (const float*)d_in[1];
    const float* qvctx[4] = { (const float*)d_in[2], (const float*)d_in[4],
                              (const float*)d_in[6], (const float*)d_in[8] };
    const int*   hits[4]  = { (const int*)d_in[3], (const int*)d_in[5],
                              (const int*)d_in[7], (const int*)d_in[9] };
    const float* qfc_w = (const float*)d_in[10];
    const float* qfc_b = (const float*)d_in[11];
    const float* pos_emb = (const float*)d_in[12];
    const float* wq = (const float*)d_in[13];
    const float* bq = (const float*)d_in[14];
    const float* wk = (const float*)d_in[15];
    const float* bk = (const float*)d_in[16];
    const float* wv = (const float*)d_in[17];
    const float* bv = (const float*)d_in[18];

    float* out = (float*)d_out;
    float* qn  = (float*)d_ws;   // 32*512 floats

    qn_kernel<<<32, 256, 0, stream>>>(qfeat, qfc_w, qfc_b, qn);

    size_t feat_off = 4096;  // after 4x merge (512) + 4x ctn_scores (512)
    for (int i = 0; i < 4; ++i) {
        const int snip = 10 << i;
        const int A = 1600 / snip;
        const int ST16 = ((snip + 15) / 16) * 16;
        const int SP = ST16 + 4;
        const size_t shmem =
            (size_t)(ST16 * 516 + 3 * ST16 * 68 + ST16 * SP + 512 + 2 * ST16 + 256) *
            sizeof(float);
        hipFuncSetAttribute(reinterpret_cast<const void*>(encode_kernel),
                            hipFuncAttributeMaxDynamicSharedMemorySize, (int)shmem);
        encode_kernel<<<512, 256, shmem, stream>>>(
            vfeats, pos_emb, wq, bq, wk, bk, wv, bv,
            hits[i], qvctx[i], qn,
            out + feat_off, out + (size_t)i * 512, out + 2048 + (size_t)i * 512,
            snip, A);
        feat_off += (size_t)512 * snip * 512;
    }
}